// SourceCriterion_86165633892925
// MI455X (gfx1250) — compile-verified
//
#include <hip/hip_runtime.h>
#include <math.h>

typedef float v2f __attribute__((ext_vector_type(2)));
typedef float v8f __attribute__((ext_vector_type(8)));

// Exact f32 full-wave (32-lane) sum via two V_WMMA_F32_16X16X4_F32 ops.
// A = {x, 0} per lane -> A(M,0)=x[M], A(M,2)=x[M+16]; B = ones (4x16).
// D(M,N) = x[M] + x[M+16] for all N.  Per-lane sum of the 8 D components
// gives S_lo on lanes 0-15 and S_hi on lanes 16-31; a second WMMA adds the
// two halves, broadcasting the exact total to every lane. Deterministic
// addition order; requires EXEC == all-ones (callers guarantee full waves).
__device__ __forceinline__ float wave_reduce_wmma(float x) {
  v2f a;   a.x = x;    a.y = 0.0f;
  v2f one; one.x = 1.0f; one.y = 1.0f;
  v8f c = {};
  v8f d = __builtin_amdgcn_wmma_f32_16x16x4_f32(false, a, false, one, (short)0, c, false, false);
  float s = ((d[0] + d[1]) + (d[2] + d[3])) + ((d[4] + d[5]) + (d[6] + d[7]));
  v2f a2; a2.x = s; a2.y = 0.0f;
  v8f d2 = __builtin_amdgcn_wmma_f32_16x16x4_f32(false, a2, false, one, (short)0, c, false, false);
  return d2[0];
}

#define QFL_BLOCKS 256   // blocks per batch for the QFL stream
#define BLK 256          // 8 waves of 32

// ---------------- QFL: streaming BCE * |sigmoid(p)-t|^2 ----------------
__global__ void qfl_kernel(const float* __restrict__ pred,
                           const float* __restrict__ tgt,
                           float* __restrict__ part_sum,
                           float* __restrict__ part_pos,
                           long perBatch) {
  const int b = blockIdx.y;
  const long base = (long)b * perBatch;
  const long stride = (long)gridDim.x * blockDim.x;
  float s = 0.0f, pc = 0.0f;
  for (long i = (long)blockIdx.x * blockDim.x + threadIdx.x; i < perBatch; i += stride) {
    __builtin_prefetch((const void*)(pred + base + i + stride), 0, 1);
    __builtin_prefetch((const void*)(tgt  + base + i + stride), 0, 1);
    float p = pred[base + i];
    float t = tgt[base + i];
    float bce = fmaxf(p, 0.0f) - p * t + log1pf(expf(-fabsf(p)));
    float sig = 1.0f / (1.0f + expf(-p));
    float df  = fabsf(sig - t);
    s  += bce * df * df;
    pc += (t > 0.0f) ? 1.0f : 0.0f;
  }
  // block reduction: per-wave WMMA reduce, then wave 0 WMMA-reduces wave totals
  __shared__ float sm0[8], sm1[8];
  float ws = wave_reduce_wmma(s);
  float wp = wave_reduce_wmma(pc);
  int lane = threadIdx.x & 31, wid = threadIdx.x >> 5;
  if (lane == 0) { sm0[wid] = ws; sm1[wid] = wp; }
  __syncthreads();
  if (wid == 0) {
    int nw = blockDim.x >> 5;
    float a0 = (lane < nw) ? sm0[lane] : 0.0f;
    float a1 = (lane < nw) ? sm1[lane] : 0.0f;
    float r0 = wave_reduce_wmma(a0);
    float r1 = wave_reduce_wmma(a1);
    if (lane == 0) {
      part_sum[(long)b * gridDim.x + blockIdx.x] = r0;
      part_pos[(long)b * gridDim.x + blockIdx.x] = r1;
    }
  }
}

// ---------------- fused DFL + GIoU over pred_reg (B,4,BINS,N) ----------------
__global__ void reg_kernel(const float* __restrict__ pred_reg,
                           const float* __restrict__ gt_dfl,
                           const float* __restrict__ anchors,
                           const float* __restrict__ strides,
                           float* __restrict__ part_dfl,
                           float* __restrict__ part_giou,
                           float* __restrict__ part_pos,
                           int N, int BINS) {
  const int b = blockIdx.y;
  const int n = blockIdx.x * blockDim.x + threadIdx.x;
  const bool valid = (n < N);
  const int nn = valid ? n : 0;

  float tk[4];
  bool mask = false;
  #pragma unroll
  for (int k = 0; k < 4; ++k) {
    tk[k] = gt_dfl[((long)b * 4 + k) * N + nn];
    mask = mask || (fabsf(tk[k]) < 1e30f);   // ~isinf
  }

  const float* pr = pred_reg + (long)b * 4 * BINS * N;
  float dfl_c = 0.0f;
  float dk[4];
  #pragma unroll
  for (int k = 0; k < 4; ++k) {
    float x[17];
    #pragma unroll
    for (int j = 0; j < 17; ++j)
      x[j] = pr[((long)(k * BINS + j)) * N + nn];
    float m = x[0];
    #pragma unroll
    for (int j = 1; j < 17; ++j) m = fmaxf(m, x[j]);
    float se = 0.0f, we = 0.0f;
    #pragma unroll
    for (int j = 0; j < 17; ++j) {
      float e = expf(x[j] - m);
      se += e;
      we += (float)j * e;
    }
    float lse = m + logf(se);
    dk[k] = we / se;                     // softmax expectation (proj dot)

    float t = mask ? tk[k] : 0.0f;
    int tl = (int)t;
    tl = max(0, min(tl, 15));
    float wr = t - (float)tl;
    float wl = 1.0f - wr;
    float xl = x[0], xr = x[1];
    #pragma unroll
    for (int j = 0; j < 17; ++j) {
      xl = (j == tl)     ? x[j] : xl;
      xr = (j == tl + 1) ? x[j] : xr;
    }
    float ce = (lse - xl) * wl + (lse - xr) * wr;
    dfl_c += mask ? ce : 0.0f;
  }

  // GIoU
  const float EPS = 1e-9f, WH = 1024.0f;
  float sN = strides[nn];
  float ax = anchors[2 * nn], ay = anchors[2 * nn + 1];
  float db[4], tb[4];
  #pragma unroll
  for (int k = 0; k < 4; ++k) {
    db[k] = dk[k] * sN;
    tb[k] = (mask ? tk[k] : 0.0f) * sN;
  }
  float px1 = fminf(fmaxf(ax - db[0], 0.0f), WH);
  float py1 = fminf(fmaxf(ay - db[1], 0.0f), WH);
  float px2 = fminf(fmaxf(ax + db[2], 0.0f), WH);
  float py2 = fminf(fmaxf(ay + db[3], 0.0f), WH);
  float tx1 = fminf(fmaxf(ax - tb[0], 0.0f), WH);
  float ty1 = fminf(fmaxf(ay - tb[1], 0.0f), WH);
  float tx2 = fminf(fmaxf(ax + tb[2], 0.0f), WH);
  float ty2 = fminf(fmaxf(ay + tb[3], 0.0f), WH);
  float iw = fmaxf(fminf(px2, tx2) - fmaxf(px1, tx1), 0.0f);
  float ih = fmaxf(fminf(py2, ty2) - fmaxf(py1, ty1), 0.0f);
  float inter = iw * ih;
  float area_p = (px2 - px1) * (py2 - py1);
  float area_t = (tx2 - tx1) * (ty2 - ty1);
  float un = area_p + area_t - inter;
  float iou = inter / fmaxf(un, EPS);
  float cw = fmaxf(px2, tx2) - fminf(px1, tx1);
  float ch = fmaxf(py2, ty2) - fminf(py1, ty1);
  float area_c = cw * ch;
  float giou = iou - (area_c - un) / fmaxf(area_c, EPS);
  float giou_c = (1.0f - giou) * (mask ? 1.0f : 0.0f);
  float pos_c  = mask ? 1.0f : 0.0f;

  if (!valid) { dfl_c = 0.0f; giou_c = 0.0f; pos_c = 0.0f; }

  __shared__ float sm0[8], sm1[8], sm2[8];
  float w0 = wave_reduce_wmma(dfl_c);
  float w1 = wave_reduce_wmma(giou_c);
  float w2 = wave_reduce_wmma(pos_c);
  int lane = threadIdx.x & 31, wid = threadIdx.x >> 5;
  if (lane == 0) { sm0[wid] = w0; sm1[wid] = w1; sm2[wid] = w2; }
  __syncthreads();
  if (wid == 0) {
    int nw = blockDim.x >> 5;
    float a0 = (lane < nw) ? sm0[lane] : 0.0f;
    float a1 = (lane < nw) ? sm1[lane] : 0.0f;
    float a2 = (lane < nw) ? sm2[lane] : 0.0f;
    float r0 = wave_reduce_wmma(a0);
    float r1 = wave_reduce_wmma(a1);
    float r2 = wave_reduce_wmma(a2);
    if (lane == 0) {
      part_dfl [(long)b * gridDim.x + blockIdx.x] = r0;
      part_giou[(long)b * gridDim.x + blockIdx.x] = r1;
      part_pos [(long)b * gridDim.x + blockIdx.x] = r2;
    }
  }
}

// ---------------- final combine: one wave, WMMA reductions ----------------
__global__ void combine_kernel(const float* __restrict__ qs,
                               const float* __restrict__ qp,
                               const float* __restrict__ ds,
                               const float* __restrict__ gs,
                               const float* __restrict__ pc,
                               int GQ, int GR, int B,
                               float* __restrict__ out) {
  int lane = threadIdx.x;  // exactly 32 threads
  float qfl_acc = 0.0f, dfl_acc = 0.0f, giou_acc = 0.0f;
  for (int b = 0; b < B; ++b) {
    float a;
    a = 0.0f; for (int g = lane; g < GQ; g += 32) a += qs[(long)b * GQ + g];
    float Sq = wave_reduce_wmma(a);
    a = 0.0f; for (int g = lane; g < GQ; g += 32) a += qp[(long)b * GQ + g];
    float Pq = wave_reduce_wmma(a);
    a = 0.0f; for (int g = lane; g < GR; g += 32) a += ds[(long)b * GR + g];
    float Sd = wave_reduce_wmma(a);
    a = 0.0f; for (int g = lane; g < GR; g += 32) a += gs[(long)b * GR + g];
    float Sg = wave_reduce_wmma(a);
    a = 0.0f; for (int g = lane; g < GR; g += 32) a += pc[(long)b * GR + g];
    float Pr = wave_reduce_wmma(a);

    qfl_acc  += Sq / (Pq + 0.001f);
    dfl_acc  += Sd / (4.0f * Pr + 0.001f);
    giou_acc += (Pr == 0.0f) ? 0.0f : (Sg / fmaxf(Pr, 1.0f));
  }
  if (lane == 0) {
    out[0] = qfl_acc  / (float)B;
    out[1] = dfl_acc  / (float)B;
    out[2] = giou_acc / (float)B;
  }
}

extern "C" void kernel_launch(void* const* d_in, const int* in_sizes, int n_in,
                              void* d_out, int out_size, void* d_ws, size_t ws_size,
                              hipStream_t stream) {
  const float* gt_qfl   = (const float*)d_in[0];  // (B, N, C)
  const float* gt_dfl   = (const float*)d_in[1];  // (B, 4, N)
  const float* anchors  = (const float*)d_in[2];  // (N, 2)
  const float* fstrides = (const float*)d_in[3];  // (N,)
  const float* pred_cls = (const float*)d_in[4];  // (B, N, C)
  const float* pred_reg = (const float*)d_in[5];  // (B, 4, BINS, N)

  const int N    = in_sizes[3];
  const int B    = in_sizes[1] / (4 * N);
  const int BINS = in_sizes[5] / (B * 4 * N);
  const long perBatch = (long)in_sizes[0] / B;    // N * C

  const int GQ = QFL_BLOCKS;
  const int GR = (N + BLK - 1) / BLK;

  float* ws = (float*)d_ws;
  float* p_qs = ws;                        // B*GQ  qfl loss partials
  float* p_qp = p_qs + (long)B * GQ;       // B*GQ  qfl pos partials
  float* p_ds = p_qp + (long)B * GQ;       // B*GR  dfl partials
  float* p_gs = p_ds + (long)B * GR;       // B*GR  giou partials
  float* p_pc = p_gs + (long)B * GR;       // B*GR  pos-count partials

  qfl_kernel<<<dim3(GQ, B), BLK, 0, stream>>>(pred_cls, gt_qfl, p_qs, p_qp, perBatch);
  reg_kernel<<<dim3(GR, B), BLK, 0, stream>>>(pred_reg, gt_dfl, anchors, fstrides,
                                              p_ds, p_gs, p_pc, N, BINS);
  combine_kernel<<<1, 32, 0, stream>>>(p_qs, p_qp, p_ds, p_gs, p_pc,
                                       GQ, GR, B, (float*)d_out);
}